// StochasticDownsampling3D_47218870453101
// MI455X (gfx1250) — compile-verified
//
#include <hip/hip_runtime.h>
#include <stdint.h>

// Output: (1, 64, 64, 64, 32) f32 = 8,388,608 floats.
// Pure gather + channel-broadcast; bandwidth-bound (~64-70 MB total traffic).
typedef float v4f __attribute__((ext_vector_type(4)));

// Address-space-qualified int typedefs matching the async-LDS builtin's
// expected parameter types ('__device__ int *' / local int*).
typedef __attribute__((address_space(1))) int gas_int;  // global
typedef __attribute__((address_space(3))) int las_int;  // LDS

__global__ __launch_bounds__(256) void sdown3d_bcast_kernel(
    const float* __restrict__ in,   // (128,128,128,32) f32, channel 0 only
    const int*   __restrict__ zi,   // 64
    const int*   __restrict__ ri,   // 64
    const int*   __restrict__ ci,   // 64
    float*       __restrict__ out)  // (64,64,64,32) f32
{
    __shared__ float g[64];

    const int tid = threadIdx.x;
    const int blk = blockIdx.x;          // 0..4095 == z'*64 + r'
    const int zp  = blk >> 6;
    const int rp  = blk & 63;

    // Block-uniform -> scalar loads
    const long z = zi[zp];
    const long r = ri[rp];

    if (tid < 64) {
        const long c = (long)ci[tid];
        const float* src = in + (((z * 128 + r) * 128 + c) * 32);  // channel 0
#if defined(__gfx1250__) && __has_builtin(__builtin_amdgcn_global_load_async_to_lds_b32)
        // CDNA5 async global->LDS DMA (tracked by ASYNCcnt), bypasses VGPRs.
        __builtin_amdgcn_global_load_async_to_lds_b32(
            (gas_int*)(uintptr_t)src,
            (las_int*)(uint32_t)(uintptr_t)&g[tid],
            /*offset=*/0, /*cpol=*/0);
#else
        g[tid] = __builtin_nontemporal_load(src);
#endif
    }

#if defined(__gfx1250__) && __has_builtin(__builtin_amdgcn_global_load_async_to_lds_b32)
#if __has_builtin(__builtin_amdgcn_s_wait_asynccnt)
    __builtin_amdgcn_s_wait_asynccnt(0);
#else
    asm volatile("s_wait_asynccnt 0" ::: "memory");
#endif
#endif
    __syncthreads();

    // Each block writes 2048 floats = 512 float4, fully coalesced NT stores.
    // float4 index q covers channels [q*4 % 32 .. +3] of spatial c' = q>>3.
    v4f* out4 = (v4f*)out + (size_t)blk * 512;

    const float v0 = g[tid >> 3];          // q  = tid        -> c' = tid>>3
    const float v1 = g[32 + (tid >> 3)];   // q2 = tid + 256  -> c' = 32 + (tid>>3)
    v4f a = {v0, v0, v0, v0};
    v4f b = {v1, v1, v1, v1};
    __builtin_nontemporal_store(a, out4 + tid);
    __builtin_nontemporal_store(b, out4 + 256 + tid);
}

extern "C" void kernel_launch(void* const* d_in, const int* in_sizes, int n_in,
                              void* d_out, int out_size, void* d_ws, size_t ws_size,
                              hipStream_t stream) {
    const float* in = (const float*)d_in[0];
    const int*   zi = (const int*)d_in[1];
    const int*   ri = (const int*)d_in[2];
    const int*   ci = (const int*)d_in[3];
    // d_in[4] = t (constant 4), unused: geometry is baked in (128 -> 64 per axis).
    float* out = (float*)d_out;

    // 4096 blocks: one (z', r') row each; 256 threads = 8 wave32 waves.
    sdown3d_bcast_kernel<<<64 * 64, 256, 0, stream>>>(in, zi, ri, ci, out);
}